// _HcSplitSinkhornModule_29566554866380
// MI455X (gfx1250) — compile-verified
//
#include <hip/hip_runtime.h>
#include <stdint.h>

// Problem constants (from reference)
#define N_TOK    131072
#define C_DIM    1024
#define G_DIM    8
#define E_DIM    128
#define INV_EPS  20.0f          // 1/0.05

// Launch geometry
#define WAVES_PER_BLOCK 4
#define BLOCK_THREADS   (WAVES_PER_BLOCK * 32)
#define GRID_BLOCKS     1024
#define TOTAL_WAVES     (WAVES_PER_BLOCK * GRID_BLOCKS)
#define TOK_PER_WAVE    (N_TOK / TOTAL_WAVES)      // 32
#define NBUF            3                           // async pipeline depth

typedef float vf4 __attribute__((ext_vector_type(4)));   // native vector for NT store

// ---- CDNA5 async global->LDS copy (ASYNCcnt-tracked), 512B per instruction ----
#define ASYNC_LD16(LDSOFF, VOFF, SBASE, IMM)                                   \
  asm volatile("global_load_async_to_lds_b128 %0, %1, %2 offset:" IMM          \
               :: "v"(LDSOFF), "v"(VOFF), "s"(SBASE) : "memory")

#define WAIT_ASYNC(N) asm volatile("s_wait_asynccnt " #N ::: "memory")
#define WAIT_DS0()    asm volatile("s_wait_dscnt 0x0" ::: "memory")

__device__ __forceinline__ void issue_row_async(uint32_t ldsoff, uint32_t voff,
                                                uint64_t sbase) {
  // one 4KB token row = 8 x (32 lanes x 16B); imm offset applies to both LDS and global
  ASYNC_LD16(ldsoff, voff, sbase, "0");
  ASYNC_LD16(ldsoff, voff, sbase, "512");
  ASYNC_LD16(ldsoff, voff, sbase, "1024");
  ASYNC_LD16(ldsoff, voff, sbase, "1536");
  ASYNC_LD16(ldsoff, voff, sbase, "2048");
  ASYNC_LD16(ldsoff, voff, sbase, "2560");
  ASYNC_LD16(ldsoff, voff, sbase, "3072");
  ASYNC_LD16(ldsoff, voff, sbase, "3584");
}

__device__ __forceinline__ float wave_max32(float x) {
#pragma unroll
  for (int off = 16; off > 0; off >>= 1) x = fmaxf(x, __shfl_xor(x, off, 32));
  return x;
}
__device__ __forceinline__ float wave_sum32(float x) {
#pragma unroll
  for (int off = 16; off > 0; off >>= 1) x += __shfl_xor(x, off, 32);
  return x;
}

// MODE 0: first column pass (u=0), accumulate exp(K - shift_g) into colsum
// MODE 1: fused row-softmax + column pass: acc += softmax_e(K+v) into colsum
// MODE 2: final output: out = softmax_e(K+v)
template <int MODE>
__global__ __launch_bounds__(BLOCK_THREADS)
void sinkhorn_pass(const float* __restrict__ mixes,
                   const float* __restrict__ hc_scale,
                   const float* __restrict__ hc_base,
                   const float* __restrict__ v_io,
                   double* __restrict__ colsum,
                   float* __restrict__ out)
{
  __shared__ double lds_cs[C_DIM];                               // 8 KB
  __shared__ float  lds_buf[WAVES_PER_BLOCK][NBUF][C_DIM];       // 48 KB

  const int tid  = threadIdx.x;
  const int lane = tid & 31;
  const int widx = tid >> 5;

  if (MODE < 2) {
    for (int i = tid; i < C_DIM; i += BLOCK_THREADS) lds_cs[i] = 0.0;
  }
  __syncthreads();

  float ksc[G_DIM], kbs[G_DIM], sh0[G_DIM], vreg[G_DIM][4];
#pragma unroll
  for (int g = 0; g < G_DIM; ++g) {
    const float sc = hc_scale[g];
    const float bs = hc_base[g];
    ksc[g] = sc * INV_EPS;
    kbs[g] = bs * INV_EPS;
    sh0[g] = (5.5f * sc + bs) * INV_EPS;   // safe upper bound on column max of K
    if (MODE >= 1) {
      const float4 vv = *reinterpret_cast<const float4*>(&v_io[g * E_DIM + lane * 4]);
      vreg[g][0] = vv.x; vreg[g][1] = vv.y; vreg[g][2] = vv.z; vreg[g][3] = vv.w;
    }
  }

  double acc[G_DIM][4];
  if (MODE < 2) {
#pragma unroll
    for (int g = 0; g < G_DIM; ++g)
#pragma unroll
      for (int j = 0; j < 4; ++j) acc[g][j] = 0.0;
  }

  const int gw = blockIdx.x * WAVES_PER_BLOCK + widx;
  const int n0 = gw * TOK_PER_WAVE;
  const uint64_t gbase  = (uint64_t)mixes;
  const uint32_t lane16 = (uint32_t)lane * 16u;

  uint32_t ldsoff[NBUF];
#pragma unroll
  for (int b = 0; b < NBUF; ++b)
    // flat pointer to LDS: low 32 bits are the wave-relative LDS byte address
    ldsoff[b] = (uint32_t)(uintptr_t)(&lds_buf[widx][b][0]) + lane16;

  // pipeline prologue: two tokens in flight
  issue_row_async(ldsoff[0], (uint32_t)(n0 + 0) * 4096u + lane16, gbase);
  issue_row_async(ldsoff[1], (uint32_t)(n0 + 1) * 4096u + lane16, gbase);

  for (int t = 0; t < TOK_PER_WAVE; ++t) {
    const int n   = n0 + t;
    const int cur = t % NBUF;

    if (t + 2 < TOK_PER_WAVE) {
      // about to overwrite the buffer consumed at t-1: make sure this wave's
      // ds reads of it have fully completed before the async engine can write
      WAIT_DS0();
      issue_row_async(ldsoff[(t + 2) % NBUF],
                      (uint32_t)(n + 2) * 4096u + lane16, gbase);
      WAIT_ASYNC(16);          // two tokens may remain outstanding
    } else if (t + 1 < TOK_PER_WAVE) {
      WAIT_ASYNC(8);
    } else {
      WAIT_ASYNC(0);
    }

    const float* __restrict__ buf = &lds_buf[widx][cur][0];

#pragma unroll
    for (int g = 0; g < G_DIM; ++g) {
      const float4 x = *reinterpret_cast<const float4*>(&buf[g * E_DIM + lane * 4]);
      const float k0 = fmaf(x.x, ksc[g], kbs[g]);
      const float k1 = fmaf(x.y, ksc[g], kbs[g]);
      const float k2 = fmaf(x.z, ksc[g], kbs[g]);
      const float k3 = fmaf(x.w, ksc[g], kbs[g]);

      if (MODE == 0) {
        acc[g][0] += (double)__expf(k0 - sh0[g]);
        acc[g][1] += (double)__expf(k1 - sh0[g]);
        acc[g][2] += (double)__expf(k2 - sh0[g]);
        acc[g][3] += (double)__expf(k3 - sh0[g]);
      } else {
        const float y0 = k0 + vreg[g][0];
        const float y1 = k1 + vreg[g][1];
        const float y2 = k2 + vreg[g][2];
        const float y3 = k3 + vreg[g][3];
        float m = fmaxf(fmaxf(y0, y1), fmaxf(y2, y3));
        m = wave_max32(m);                       // row max over E=128
        const float e0 = __expf(y0 - m);
        const float e1 = __expf(y1 - m);
        const float e2 = __expf(y2 - m);
        const float e3 = __expf(y3 - m);
        float s = (e0 + e1) + (e2 + e3);
        s = wave_sum32(s);                       // row sum over E=128
        const float rs = 1.0f / s;               // softmax = exp(K+u+v), terms <= 1
        if (MODE == 1) {
          acc[g][0] += (double)(e0 * rs);
          acc[g][1] += (double)(e1 * rs);
          acc[g][2] += (double)(e2 * rs);
          acc[g][3] += (double)(e3 * rs);
        } else {
          vf4 o;
          o.x = e0 * rs; o.y = e1 * rs; o.z = e2 * rs; o.w = e3 * rs;
          // streaming output: non-temporal store keeps L2 for the input stream
          __builtin_nontemporal_store(
              o, reinterpret_cast<vf4*>(&out[(size_t)n * C_DIM + g * E_DIM + lane * 4]));
        }
      }
    }
  }

  if (MODE < 2) {
#pragma unroll
    for (int g = 0; g < G_DIM; ++g)
#pragma unroll
      for (int j = 0; j < 4; ++j)
        atomicAdd(&lds_cs[g * E_DIM + lane * 4 + j], acc[g][j]);   // ds_add_f64
    __syncthreads();
    for (int i = tid; i < C_DIM; i += BLOCK_THREADS)
      atomicAdd(&colsum[i], lds_cs[i]);                            // global_atomic_add_f64
  }
}

// v_new = v_old - log(colsum)   (FIRST: v = -(shift_g + log(colsum)));  reset colsum
template <bool FIRST>
__global__ __launch_bounds__(1024)
void sinkhorn_colfin(double* __restrict__ colsum, float* __restrict__ v,
                     const float* __restrict__ hc_scale,
                     const float* __restrict__ hc_base)
{
  const int i = threadIdx.x;          // 0..1023 -> (g,e)
  const int g = i >> 7;
  const double cs = colsum[i];
  const float shift = FIRST ? (5.5f * hc_scale[g] + hc_base[g]) * INV_EPS
                            : -v[i];
  v[i] = -(shift + (float)log(cs));
  colsum[i] = 0.0;
}

extern "C" void kernel_launch(void* const* d_in, const int* in_sizes, int n_in,
                              void* d_out, int out_size, void* d_ws, size_t ws_size,
                              hipStream_t stream) {
  (void)in_sizes; (void)n_in; (void)out_size; (void)ws_size;
  const float* mixes = (const float*)d_in[0];
  const float* sc    = (const float*)d_in[1];
  const float* bs    = (const float*)d_in[2];
  float*       out   = (float*)d_out;

  double* colsum = (double*)d_ws;                                   // 8 KB
  float*  v      = (float*)((char*)d_ws + C_DIM * sizeof(double));  // 4 KB

  // ws is poisoned by the harness; zero our state every call (graph-capture safe)
  (void)hipMemsetAsync(d_ws, 0, C_DIM * sizeof(double) + C_DIM * sizeof(float), stream);

  dim3 grid(GRID_BLOCKS), block(BLOCK_THREADS);

  // iteration 1 column pass (u = 0)
  sinkhorn_pass<0><<<grid, block, 0, stream>>>(mixes, sc, bs, v, colsum, nullptr);
  sinkhorn_colfin<true><<<1, 1024, 0, stream>>>(colsum, v, sc, bs);

  // iterations 2..8: row-norm of (t-1) fused with column pass of t
  for (int t = 2; t <= 8; ++t) {
    sinkhorn_pass<1><<<grid, block, 0, stream>>>(mixes, sc, bs, v, colsum, nullptr);
    sinkhorn_colfin<false><<<1, 1024, 0, stream>>>(colsum, v, sc, bs);
  }

  // final row-norm (iteration 8) + exp -> output
  sinkhorn_pass<2><<<grid, block, 0, stream>>>(mixes, sc, bs, v, nullptr, out);
}